// GCNConv_simple_37237366456393
// MI455X (gfx1250) — compile-verified
//
#include <hip/hip_runtime.h>
#include <hip/hip_bf16.h>
#include <stdint.h>

#define IN_DIM   128
#define H1       256
#define HID      128
#define OUT_DIM  64
#define DROP_P   0.2f
#define INV_KEEP 1.25f

typedef float v2f __attribute__((ext_vector_type(2)));
typedef float v8f __attribute__((ext_vector_type(8)));

// ---------------- degree / norm ----------------

__global__ void deg_init_kernel(float* deg, int n) {
  int i = blockIdx.x * blockDim.x + threadIdx.x;
  if (i < n) deg[i] = 1.0f;  // self-loop contribution
}

__global__ void deg_count_kernel(const int* __restrict__ tgt, float* deg, int ne) {
  int e = blockIdx.x * blockDim.x + threadIdx.x;
  if (e < ne) unsafeAtomicAdd(&deg[tgt[e]], 1.0f);
}

__global__ void dinv_kernel(float* deg, int n) {
  int i = blockIdx.x * blockDim.x + threadIdx.x;
  if (i < n) {
    float d = deg[i];
    deg[i] = (d > 0.0f) ? rsqrtf(d) : 0.0f;  // in place: deg -> dinv
  }
}

// norm[e] = dinv[src[e]] * dinv[tgt[e]]
__global__ void norm_kernel(const int* __restrict__ src, const int* __restrict__ tgt,
                            const float* __restrict__ dinv, float* __restrict__ norm,
                            int ne) {
  int e = blockIdx.x * blockDim.x + threadIdx.x;
  if (e < ne) norm[e] = dinv[src[e]] * dinv[tgt[e]];
}

// ---------------- weight transpose (runs once per launch; weights are tiny) ----------------
// WT[n*K + k] = W[k*N + n]  -> makes B fragments contiguous b64 loads in the GEMM.
__global__ void transpose_kernel(const float* __restrict__ W, float* __restrict__ WT,
                                 int K, int N) {
  int t = blockIdx.x * blockDim.x + threadIdx.x;
  if (t >= K * N) return;
  int k = t / N;
  int nn = t - k * N;
  WT[(size_t)nn * K + k] = W[t];
}

// ---------------- f32 WMMA GEMM: C = A(MxK) @ B(KxN), B given transposed (NxK) ----------------
// One wave32 per 16x64 strip of C: one A fragment feeds 4 WMMAs.
// A-frag (16x4 f32): lanes 0-15 row M=r hold {K0,K1}; lanes 16-31 hold {K2,K3}.
// B-frag (4x16 f32): VGPR0 = rows {K0|K2} striped over lanes, VGPR1 = {K1|K3};
//                    from BT this is the contiguous pair BT[n*K + k .. k+1].
// C/D (16x16 f32): VGPR i: lanes 0-15 -> M=i, lanes 16-31 -> M=8+i, N=lane%16.
// Fused epilogues:
//   WITH_BIAS: C += bias (final layer)
//   WITH_AGG : also write agg = C * dinv[row]^2  (self-loop init for scatter)

template <int K, int N, bool WITH_BIAS, bool WITH_AGG>
__global__ void wmma_gemm_f32(const float* __restrict__ A,
                              const float* __restrict__ BT,
                              float* __restrict__ C,
                              const float* __restrict__ bias,
                              const float* __restrict__ dinv,
                              float* __restrict__ agg,
                              int M) {
  constexpr int NG = N / 64;  // 64-wide strips
  int wave = (int)((blockIdx.x * blockDim.x + threadIdx.x) >> 5);
  int lane = threadIdx.x & 31;
  int total = (M >> 4) * NG;
  if (wave >= total) return;

  int mt = wave / NG;
  int ng = wave - mt * NG;
  int m0 = mt << 4;
  int n0 = ng << 6;

  int half = lane >> 4;  // 0 -> K pair {0,1}, 1 -> K pair {2,3}
  int r    = lane & 15;

  const float* arow = A + (size_t)(m0 + r) * K + 2 * half;
  const float* btr  = BT + (size_t)(n0 + r) * K + 2 * half;

  v8f acc[4] = {};
#pragma unroll 4
  for (int k = 0; k < K; k += 4) {
    v2f a = *(const v2f*)(arow + k);  // 8B-aligned contiguous {k+2h, k+2h+1}
#pragma unroll
    for (int j = 0; j < 4; ++j) {
      v2f b = *(const v2f*)(btr + (size_t)(16 * j) * K + k);
      acc[j] = __builtin_amdgcn_wmma_f32_16x16x4_f32(
          false, a, false, b, (short)0, acc[j], false, false);
    }
  }

  float bv[4];
#pragma unroll
  for (int j = 0; j < 4; ++j) bv[j] = WITH_BIAS ? bias[n0 + 16 * j + r] : 0.0f;

  int row0 = m0 + 8 * half;
#pragma unroll
  for (int i = 0; i < 8; ++i) {
    int row = row0 + i;
    float s = 0.0f;
    if (WITH_AGG) { s = dinv[row]; s = s * s; }
    float* crow = C + (size_t)row * N + n0 + r;
#pragma unroll
    for (int j = 0; j < 4; ++j) {
      float v = acc[j][i];
      crow[16 * j] = v + bv[j];
      if (WITH_AGG) (agg + (size_t)row * N + n0 + r)[16 * j] = v * s;
    }
  }
}

// ---------------- edge scatter: agg[tgt,:] += hw[src,:] * norm[e] ----------------

__global__ void scatter_kernel(const float* __restrict__ hw,
                               const int* __restrict__ src,
                               const int* __restrict__ tgt,
                               const float* __restrict__ norm,
                               float* __restrict__ agg, int ne, int D) {
  int chunks = D >> 2;
  int t = blockIdx.x * blockDim.x + threadIdx.x;
  if (t >= ne * chunks) return;
  int e = t / chunks;
  int c = t - e * chunks;
  int s = src[e];
  int g = tgt[e];
  float nm = norm[e];
  float4 v = ((const float4*)(hw + (size_t)s * D))[c];
  float* out = agg + (size_t)g * D + (c << 2);
  unsafeAtomicAdd(out + 0, v.x * nm);
  unsafeAtomicAdd(out + 1, v.y * nm);
  unsafeAtomicAdd(out + 2, v.z * nm);
  unsafeAtomicAdd(out + 3, v.w * nm);
}

// ---------------- bias + dropout + relu (in place) ----------------

__device__ __forceinline__ float hash_u01(uint32_t x) {
  x ^= x >> 16; x *= 0x7feb352du;
  x ^= x >> 15; x *= 0x846ca68bu;
  x ^= x >> 16;
  return (float)x * 2.3283064365386963e-10f;  // [0,1)
}

__global__ void post_kernel(float* __restrict__ h, const float* __restrict__ bias,
                            int n, int D, uint32_t seed) {
  int t = blockIdx.x * blockDim.x + threadIdx.x;
  if (t >= n * D) return;
  int d = t % D;
  float z = h[t] + bias[d];
  float u = hash_u01(seed ^ ((uint32_t)t * 0x9E3779B9u));
  h[t] = (u >= DROP_P) ? fmaxf(z, 0.0f) * INV_KEEP : 0.0f;
}

// ---------------- launch ----------------

static inline int cdiv(long long a, int b) { return (int)((a + b - 1) / b); }

extern "C" void kernel_launch(void* const* d_in, const int* in_sizes, int n_in,
                              void* d_out, int out_size, void* d_ws, size_t ws_size,
                              hipStream_t stream) {
  const float* x  = (const float*)d_in[0];
  const float* W1 = (const float*)d_in[1];
  const float* b1 = (const float*)d_in[2];
  const float* W2 = (const float*)d_in[3];
  const float* b2 = (const float*)d_in[4];
  const float* W3 = (const float*)d_in[5];
  const float* b3 = (const float*)d_in[6];
  const int*   ei = (const int*)d_in[7];

  const int n  = in_sizes[0] / IN_DIM;   // 50000
  const int ne = in_sizes[7] / 2;        // 800000
  const int* src = ei;                   // edge_index[0]
  const int* tgt = ei + ne;              // edge_index[1]

  // workspace layout
  char* ws = (char*)d_ws;
  size_t off = 0;
  auto alloc = [&](size_t elems) {
    float* p = (float*)(ws + off);
    off = (off + elems * 4 + 255) & ~(size_t)255;
    return p;
  };
  float* dinv = alloc((size_t)n);
  float* norm = alloc((size_t)ne);
  float* w1t  = alloc((size_t)IN_DIM * H1);
  float* w2t  = alloc((size_t)H1 * HID);
  float* w3t  = alloc((size_t)HID * OUT_DIM);
  float* bufA = alloc((size_t)n * H1);   // hw1 (n x 256), later hw2 (n x 128)
  float* bufB = alloc((size_t)n * H1);   // agg1 / h1 (n x 256)
  float* bufC = alloc((size_t)n * HID);  // agg2 / h2 (n x 128)

  const int B = 256;

  // weight transposes (tiny, once per launch)
  transpose_kernel<<<cdiv((long long)IN_DIM * H1, B), B, 0, stream>>>(W1, w1t, IN_DIM, H1);
  transpose_kernel<<<cdiv((long long)H1 * HID, B), B, 0, stream>>>(W2, w2t, H1, HID);
  transpose_kernel<<<cdiv((long long)HID * OUT_DIM, B), B, 0, stream>>>(W3, w3t, HID, OUT_DIM);

  // degree -> dinv -> per-edge norm
  deg_init_kernel<<<cdiv(n, B), B, 0, stream>>>(dinv, n);
  deg_count_kernel<<<cdiv(ne, B), B, 0, stream>>>(tgt, dinv, ne);
  dinv_kernel<<<cdiv(n, B), B, 0, stream>>>(dinv, n);
  norm_kernel<<<cdiv(ne, B), B, 0, stream>>>(src, tgt, dinv, norm, ne);

  // layer 1: hw1 = x @ W1 ; agg1 = hw1 * dinv^2 (fused self-loop init)
  {
    long long waves = (long long)(n >> 4) * (H1 / 64);
    wmma_gemm_f32<IN_DIM, H1, false, true>
        <<<cdiv(waves * 32, B), B, 0, stream>>>(x, w1t, bufA, nullptr, dinv, bufB, n);
  }
  scatter_kernel<<<cdiv((long long)ne * (H1 >> 2), B), B, 0, stream>>>(
      bufA, src, tgt, norm, bufB, ne, H1);
  post_kernel<<<cdiv((long long)n * H1, B), B, 0, stream>>>(bufB, b1, n, H1, 0x13371337u);

  // layer 2: hw2 = h1 @ W2 ; agg2 = hw2 * dinv^2
  {
    long long waves = (long long)(n >> 4) * (HID / 64);
    wmma_gemm_f32<H1, HID, false, true>
        <<<cdiv(waves * 32, B), B, 0, stream>>>(bufB, w2t, bufA, nullptr, dinv, bufC, n);
  }
  scatter_kernel<<<cdiv((long long)ne * (HID >> 2), B), B, 0, stream>>>(
      bufA, src, tgt, norm, bufC, ne, HID);
  post_kernel<<<cdiv((long long)n * HID, B), B, 0, stream>>>(bufC, b2, n, HID, 0x2A2A2A2Au);

  // output: out = h2 @ W3 + b3
  {
    long long waves = (long long)(n >> 4) * (OUT_DIM / 64);
    wmma_gemm_f32<HID, OUT_DIM, true, false>
        <<<cdiv(waves * 32, B), B, 0, stream>>>(bufC, w3t, (float*)d_out, b3, nullptr,
                                                nullptr, n);
  }
}